// Net_15951508537600
// MI455X (gfx1250) — compile-verified
//
#include <hip/hip_runtime.h>
#include <hip/hip_bf16.h>
#include <math.h>

typedef float v2f __attribute__((ext_vector_type(2)));
typedef float v8f __attribute__((ext_vector_type(8)));

#define NB      2048
#define NNODE   116
#define NCH     24
#define NTHREADS 512
#define NWAVES   16

// LDS layout (floats)
#define SA_LEN    (128*120)   // A, stride 120
#define SX_LEN    (128*32)    // x, stride 32
#define BUF_LEN   (128*80)    // scratch (m/gi) and (agg/gh), stride 32 or 80
#define SWC_LEN   (24*32)
#define SWT_LEN   (24*80)
#define SMEM_FLOATS (SA_LEN + SX_LEN + 2*BUF_LEN + SWC_LEN + 2*SWT_LEN + 72 + 72 + 128 + 128 + 128 + 384)

__device__ __forceinline__ float sigf(float x) { return 1.0f / (1.0f + expf(-x)); }

// Block-level f32 GEMM: D[M,N] = A[M,K] * B[K,N], all in LDS, via V_WMMA_F32_16X16X4_F32.
// Requirements: K % 4 == 0; B rows in [actualK, K) zeroed by caller if K was padded.
// Stores are UNCONDITIONAL: D must have capacity for ceil16(M) rows; rows >= M get
// garbage (finite) values that callers must never consume.
__device__ __forceinline__ void block_gemm(const float* Am, int lda,
                                           const float* Bm, int ldb,
                                           float* Dm, int ldd,
                                           int M, int N, int K) {
  const int lane = threadIdx.x & 31;
  const int wave = threadIdx.x >> 5;
  const int r    = lane & 15;
  const int hi   = lane >> 4;          // 0: lanes 0-15, 1: lanes 16-31
  const int tm16 = (M + 15) >> 4;
  const int tn16 = (N + 15) >> 4;
  const int nT   = tm16 * tn16;
  for (int t = wave; t < nT; t += NWAVES) {
    const int tm = (t / tn16) << 4;
    const int tn = (t % tn16) << 4;
    const int arow = tm + r;
    const int bcol = tn + r;
    v8f acc = {0.f, 0.f, 0.f, 0.f, 0.f, 0.f, 0.f, 0.f};
    for (int k = 0; k < K; k += 4) {
      const int ka = k + (hi << 1);     // ISA 7.12.2: A 16x4, halves hold K pairs {k,k+1}/{k+2,k+3}
      v2f a, b;
      a.x = Am[arow * lda + ka];
      a.y = Am[arow * lda + ka + 1];
      b.x = Bm[ka * ldb + bcol];        // B 4x16 mirrored: rows striped across lanes per VGPR
      b.y = Bm[(ka + 1) * ldb + bcol];
      acc = __builtin_amdgcn_wmma_f32_16x16x4_f32(false, a, false, b, (short)0, acc, false, false);
    }
    const int drow = tm + (hi << 3);    // C/D: VGPR rr -> row rr (lanes 0-15) / rr+8 (lanes 16-31)
    const int dcol = tn + r;
#pragma unroll
    for (int rr = 0; rr < 8; ++rr) {
      Dm[(drow + rr) * ldd + dcol] = acc[rr];
    }
  }
}

__global__ __launch_bounds__(NTHREADS, 1)
void gnn_pipeline(const float* __restrict__ gX,      // [B,116,24]
                  const float* __restrict__ gA,      // [B,116,116]
                  const float* __restrict__ gConvW,  // [6,2,24,24]
                  const float* __restrict__ gWih,    // [6,72,24]
                  const float* __restrict__ gWhh,    // [6,72,24]
                  const float* __restrict__ gBih,    // [6,72]
                  const float* __restrict__ gBhh,    // [6,72]
                  const float* __restrict__ gPoolW,  // [3,24]
                  const float* __restrict__ gS2Wih,  // [96,48]
                  const float* __restrict__ gS2Whh,  // [96,24]
                  const float* __restrict__ gS2bih,  // [96]
                  const float* __restrict__ gS2bhh,  // [96]
                  const float* __restrict__ gLinW,   // [96,32]
                  const float* __restrict__ gLinB,   // [32]
                  float* __restrict__ gOut) {
  extern __shared__ __align__(16) float smem[];
  float* sA    = smem;                    // [128][120]
  float* sX    = sA    + SA_LEN;          // [128][32]
  float* buf1  = sX    + SX_LEN;          // m (stride 32) -> gi (stride 80); pool temp A
  float* buf2  = buf1  + BUF_LEN;         // agg (stride 32) -> gh (stride 80); pool temp x
  float* sWc   = buf2  + BUF_LEN;         // [24][32]
  float* sWihT = sWc   + SWC_LEN;         // [24][80]  (transposed)
  float* sWhhT = sWihT + SWT_LEN;         // [24][80]
  float* sBih  = sWhhT + SWT_LEN;         // [72]
  float* sBhh  = sBih  + 72;              // [72]
  float* sScore= sBhh  + 72;              // [128]
  float* sVals = sScore+ 128;             // [128]
  int*   sPerm = (int*)(sVals + 128);     // [128]
  float* sMisc = (float*)(sPerm + 128);   // [384]: feat[0..95] h[96..119] c[120..143] q[144..191] g[192..287] e[288..] scalars[380..]

  const int b   = blockIdx.x;
  const int tid = threadIdx.x;

  // ---- init: everything finite/zero (vectorized; sA..buf2 contiguous) ----
  {
    const float4 z4 = make_float4(0.f, 0.f, 0.f, 0.f);
    float4* s4 = reinterpret_cast<float4*>(smem);
    for (int i = tid; i < (SA_LEN + SX_LEN + 2 * BUF_LEN) / 4; i += NTHREADS) s4[i] = z4;
  }
  __syncthreads();
  // ---- vectorized global -> LDS staging (116 and 24 are multiples of 4: no row straddle) ----
  {
    const float4* gA4 = reinterpret_cast<const float4*>(gA + (size_t)b * (NNODE * NNODE));
    for (int i4 = tid; i4 < (NNODE * NNODE) / 4; i4 += NTHREADS) {
      const float4 v = gA4[i4];
      const int r = i4 / 29, c = (i4 % 29) * 4;   // 116/4 = 29 quads per row
      *reinterpret_cast<float4*>(&sA[r * 120 + c]) = v;
    }
    const float4* gX4 = reinterpret_cast<const float4*>(gX + (size_t)b * (NNODE * NCH));
    for (int i4 = tid; i4 < (NNODE * NCH) / 4; i4 += NTHREADS) {
      const float4 v = gX4[i4];
      const int r = i4 / 6, c = (i4 % 6) * 4;     // 24/4 = 6 quads per row
      *reinterpret_cast<float4*>(&sX[r * 32 + c]) = v;
    }
  }
  __syncthreads();

  int n = NNODE;

  for (int layer = 0; layer < 6; ++layer) {
    // GRU weights (transposed for use as GEMM B operand) + biases
    for (int i = tid; i < 72 * 24; i += NTHREADS) {
      const int o = i / 24, c = i % 24;
      sWihT[c * 80 + o] = gWih[layer * 72 * 24 + i];
      sWhhT[c * 80 + o] = gWhh[layer * 72 * 24 + i];
    }
    if (tid < 72) { sBih[tid] = gBih[layer * 72 + tid]; sBhh[tid] = gBhh[layer * 72 + tid]; }
    __syncthreads();

    const int Kpad = (n + 3) & ~3;
    for (int it = 0; it < 2; ++it) {
      for (int i = tid; i < 24 * 24; i += NTHREADS)
        sWc[(i / 24) * 32 + (i % 24)] = gConvW[(layer * 2 + it) * 576 + i];
      __syncthreads();

      // m = x @ W          [n,24] = [n,24][24,24]
      block_gemm(sX, 32, sWc, 32, buf1, 32, n, 24, 24);
      __syncthreads();
      // zero K-padding rows of m (B operand of next GEMM); also cleans garbage rows
      for (int i = tid; i < (Kpad - n) * 32; i += NTHREADS) buf1[n * 32 + i] = 0.f;
      __syncthreads();
      // agg = A @ m        [n,24] = [n,n][n,24]
      block_gemm(sA, 120, buf1, 32, buf2, 32, n, 24, Kpad);
      __syncthreads();
      // gi = agg @ Wih^T   [n,72] = [n,24][24,72]   (overwrites m)
      block_gemm(buf2, 32, sWihT, 80, buf1, 80, n, 72, 24);
      __syncthreads();
      // gh = x @ Whh^T     [n,72]                    (overwrites agg)
      block_gemm(sX, 32, sWhhT, 80, buf2, 80, n, 72, 24);
      __syncthreads();
      // GRU elementwise update
      for (int i = tid; i < n * 24; i += NTHREADS) {
        const int j = i / 24, c = i % 24;
        const float ir = buf1[j * 80 + c]      + sBih[c];
        const float iz = buf1[j * 80 + 24 + c] + sBih[24 + c];
        const float in = buf1[j * 80 + 48 + c] + sBih[48 + c];
        const float hr = buf2[j * 80 + c]      + sBhh[c];
        const float hz = buf2[j * 80 + 24 + c] + sBhh[24 + c];
        const float hn = buf2[j * 80 + 48 + c] + sBhh[48 + c];
        const float r_ = sigf(ir + hr);
        const float z_ = sigf(iz + hz);
        const float n_ = tanhf(in + r_ * hn);
        const float h0 = sX[j * 32 + c];
        sX[j * 32 + c] = (1.f - z_) * n_ + z_ * h0;
      }
      __syncthreads();
    }

    // ReLU after each gated conv
    for (int i = tid; i < n * 24; i += NTHREADS) {
      const int j = i / 24, c = i % 24;
      const float v = sX[j * 32 + c];
      sX[j * 32 + c] = v > 0.f ? v : 0.f;
    }
    __syncthreads();

    if (layer < 3) {
      // ---- TopK pooling ----
      const float ratio = (layer == 2) ? 0.3f : 0.8f;
      const int kkeep = (int)ceilf(ratio * (float)n);
      if (tid == 0) {
        float s = 0.f;
        for (int c = 0; c < 24; ++c) { const float w = gPoolW[layer * 24 + c]; s += w * w; }
        sMisc[380] = sqrtf(s);
      }
      __syncthreads();
      const float invn = 1.f / sMisc[380];
      if (tid < n) {
        float d = 0.f;
        for (int c = 0; c < 24; ++c) d += sX[tid * 32 + c] * gPoolW[layer * 24 + c];
        sScore[tid] = tanhf(d * invn);
      }
      __syncthreads();
      // stable descending rank (matches lax.top_k tie-break)
      if (tid < n) {
        const float sj = sScore[tid];
        int cnt = 0;
        for (int i2 = 0; i2 < n; ++i2) {
          const float si = sScore[i2];
          cnt += (si > sj) || (si == sj && i2 < tid);
        }
        if (cnt < kkeep) { sPerm[cnt] = tid; sVals[cnt] = sj; }
      }
      __syncthreads();
      // gather x (scaled by score) and A through scratch
      for (int i = tid; i < kkeep * 24; i += NTHREADS) {
        const int p = i / 24, c = i % 24;
        buf2[p * 32 + c] = sX[sPerm[p] * 32 + c] * sVals[p];
      }
      for (int i = tid; i < kkeep * kkeep; i += NTHREADS) {
        const int p = i / kkeep, q = i % kkeep;
        buf1[p * 96 + q] = sA[sPerm[p] * 120 + sPerm[q]];
      }
      __syncthreads();
      for (int i = tid; i < kkeep * 24; i += NTHREADS) {
        const int p = i / 24, c = i % 24;
        sX[p * 32 + c] = buf2[p * 32 + c];
      }
      for (int i = tid; i < kkeep * kkeep; i += NTHREADS) {
        const int p = i / kkeep, q = i % kkeep;
        sA[p * 120 + q] = buf1[p * 96 + q];
      }
      n = kkeep;
      __syncthreads();
    }
  }

  // ---- global max / mean pool ----
  if (tid < 24) {
    float mx = -1e30f, sm = 0.f;
    for (int j = 0; j < n; ++j) { const float v = sX[j * 32 + tid]; mx = fmaxf(mx, v); sm += v; }
    sMisc[tid]      = mx;               // feat[0..23]  = gmp
    sMisc[24 + tid] = sm / (float)n;    // feat[24..47] = gap
  }
  // ---- Set2Set weights into free scratch ----
  for (int i = tid; i < 96 * 48; i += NTHREADS) buf1[i] = gS2Wih[i];
  for (int i = tid; i < 96 * 24; i += NTHREADS) buf1[4608 + i] = gS2Whh[i];
  if (tid < 96) { buf1[6912 + tid] = gS2bih[tid]; buf1[7008 + tid] = gS2bhh[tid]; }
  if (tid < 24) { sMisc[96 + tid] = 0.f; sMisc[120 + tid] = 0.f; }
  if (tid < 48) sMisc[144 + tid] = 0.f;
  __syncthreads();

  for (int step = 0; step < 2; ++step) {
    if (tid < 96) {
      float g = buf1[6912 + tid] + buf1[7008 + tid];
      for (int p = 0; p < 48; ++p) g += sMisc[144 + p] * buf1[tid * 48 + p];
      for (int p = 0; p < 24; ++p) g += sMisc[96 + p]  * buf1[4608 + tid * 24 + p];
      sMisc[192 + tid] = g;
    }
    __syncthreads();
    if (tid < 24) {
      const float gi_ = sMisc[192 + tid];
      const float gf  = sMisc[192 + 24 + tid];
      const float gg  = sMisc[192 + 48 + tid];
      const float go  = sMisc[192 + 72 + tid];
      float c_ = sMisc[120 + tid];
      c_ = sigf(gf) * c_ + sigf(gi_) * tanhf(gg);
      sMisc[120 + tid] = c_;
      sMisc[96 + tid]  = sigf(go) * tanhf(c_);
    }
    __syncthreads();
    if (tid < n) {
      float e = 0.f;
      for (int c = 0; c < 24; ++c) e += sX[tid * 32 + c] * sMisc[96 + c];
      sMisc[288 + tid] = e;
    }
    __syncthreads();
    if (tid == 0) {
      float mx = sMisc[288];
      for (int j = 1; j < n; ++j) mx = fmaxf(mx, sMisc[288 + j]);
      float sm = 0.f;
      for (int j = 0; j < n; ++j) { const float a = expf(sMisc[288 + j] - mx); sMisc[288 + j] = a; sm += a; }
      sMisc[381] = sm;
    }
    __syncthreads();
    if (tid < 24) {
      const float inv = 1.f / sMisc[381];
      float r_ = 0.f;
      for (int j = 0; j < n; ++j) r_ += sMisc[288 + j] * sX[j * 32 + tid];
      sMisc[144 + tid]      = sMisc[96 + tid];  // q_star = [h, r]
      sMisc[144 + 24 + tid] = r_ * inv;
    }
    __syncthreads();
  }
  if (tid < 48) sMisc[48 + tid] = sMisc[144 + tid];  // feat[48..95] = q_star
  __syncthreads();

  // ---- final linear + relu ----
  if (tid < 32) {
    float acc = gLinB[tid];
    for (int p = 0; p < 96; ++p) acc += sMisc[p] * gLinW[p * 32 + tid];
    gOut[(size_t)b * 32 + tid] = fmaxf(acc, 0.f);
  }
  // ---- indices output: tile(arange(116), B) ----
  int* oi = reinterpret_cast<int*>(gOut) + (size_t)NB * 32;
  for (int t = tid; t < NNODE; t += NTHREADS) oi[(size_t)b * NNODE + t] = t;
}

extern "C" void kernel_launch(void* const* d_in, const int* in_sizes, int n_in,
                              void* d_out, int out_size, void* d_ws, size_t ws_size,
                              hipStream_t stream) {
  (void)in_sizes; (void)n_in; (void)out_size; (void)d_ws; (void)ws_size;
  const float* x     = (const float*)d_in[0];
  const float* A     = (const float*)d_in[1];
  const float* convW = (const float*)d_in[2];
  const float* Wih   = (const float*)d_in[3];
  const float* Whh   = (const float*)d_in[4];
  const float* bih   = (const float*)d_in[5];
  const float* bhh   = (const float*)d_in[6];
  const float* poolW = (const float*)d_in[7];
  const float* sWih  = (const float*)d_in[8];
  const float* sWhh  = (const float*)d_in[9];
  const float* sbih  = (const float*)d_in[10];
  const float* sbhh  = (const float*)d_in[11];
  const float* linW  = (const float*)d_in[12];
  const float* linB  = (const float*)d_in[13];
  float* out = (float*)d_out;

  const size_t smem_bytes = (size_t)SMEM_FLOATS * sizeof(float);
  gnn_pipeline<<<NB, NTHREADS, smem_bytes, stream>>>(
      x, A, convW, Wih, Whh, bih, bhh, poolW, sWih, sWhh, sbih, sbhh, linW, linB, out);
}